// DWSA_Loss_37778532335807
// MI455X (gfx1250) — compile-verified
//
#include <hip/hip_runtime.h>
#include <math.h>

// ---------------- problem constants ----------------
#define NROW 2048
#define NCOLP 513          // columns incl. timestamp
#define NDIM 512           // feature dim after dropping timestamp
#define EXP2F 7.3890560989306495f   // exp(THRESHOLD=2)
#define INV_GAMMA 1000.0f  // 1/BETA

typedef __attribute__((ext_vector_type(2))) float v2f;
typedef __attribute__((ext_vector_type(8))) float v8f;

// ---------------- stage 1: stable argsort by timestamp (rank counting) ----
__global__ __launch_bounds__(256) void rank_kernel(const float* __restrict__ X,
                                                   int* __restrict__ perm) {
    __shared__ float keys[NROW];
    const int tid = threadIdx.x;
    for (int j = tid; j < NROW; j += 256) keys[j] = X[(size_t)j * NCOLP + NDIM];
    __syncthreads();
    const int i = blockIdx.x * 256 + tid;
    const float ki = keys[i];
    int cnt = 0;
    #pragma unroll 8
    for (int j = 0; j < NROW; ++j) {
        const float kj = keys[j];
        cnt += (kj < ki) || (kj == ki && j < i);   // stable tie-break == jnp.argsort
    }
    perm[cnt] = i;                                  // scatter: sorted row cnt <- row i
}

// ---------------- stage 2: gather-permute + L2 normalize -----------------
__global__ __launch_bounds__(128) void norm_permute_kernel(const float* __restrict__ X,
                                                           const int* __restrict__ perm,
                                                           float* __restrict__ Y) {
    __shared__ float red[4];
    const int r = blockIdx.x;
    const int tid = threadIdx.x;
    const float* src = X + (size_t)perm[r] * NCOLP;
    const float v0 = src[tid], v1 = src[tid + 128], v2 = src[tid + 256], v3 = src[tid + 384];
    float ss = v0 * v0 + v1 * v1 + v2 * v2 + v3 * v3;
    #pragma unroll
    for (int d = 16; d; d >>= 1) ss += __shfl_xor(ss, d, 32);
    if ((tid & 31) == 0) red[tid >> 5] = ss;
    __syncthreads();
    const float tot = red[0] + red[1] + red[2] + red[3];
    const float scale = 1.0f / sqrtf(tot + 1e-10f);
    float* dst = Y + (size_t)r * NDIM;
    dst[tid] = v0 * scale; dst[tid + 128] = v1 * scale;
    dst[tid + 256] = v2 * scale; dst[tid + 384] = v3 * scale;
}

// ---------------- stage 3: fp32 WMMA GEMM, fused E = exp(1 - A.B^T) ------
// block: 8 waves arranged 2(M) x 4(N) -> 32x64 output tile per block.
// Per wave: one 16x16 tile, K-loop of 512 in steps of 4 (V_WMMA_F32_16X16X4_F32).
__global__ __launch_bounds__(256) void gemm_exp_kernel(const float* __restrict__ An,
                                                       const float* __restrict__ Bn,
                                                       float* __restrict__ E) {
    const int tid = threadIdx.x;
    const int lane = tid & 31;
    const int wv = tid >> 5;
    const int mbase = blockIdx.y * 32 + (wv >> 2) * 16;
    const int nbase = blockIdx.x * 64 + (wv & 3) * 16;
    const int l15 = lane & 15;
    const int lh = lane >> 4;          // half-wave selects K pair {0,1} vs {2,3}
    // A: lane holds row M=l15, K = 4k + 2*lh + {0,1}
    const float2* ap = reinterpret_cast<const float2*>(An + (size_t)(mbase + l15) * NDIM) + lh;
    // B^T: lane holds col N=l15 (row of Bn), same K mapping
    const float2* bp = reinterpret_cast<const float2*>(Bn + (size_t)(nbase + l15) * NDIM) + lh;

    v8f acc = {};
    #pragma unroll 4
    for (int k = 0; k < NDIM / 4; ++k) {
        const float2 af = ap[2 * k];
        const float2 bf = bp[2 * k];
        v2f a; a.x = af.x; a.y = af.y;
        v2f b; b.x = bf.x; b.y = bf.y;
        acc = __builtin_amdgcn_wmma_f32_16x16x4_f32(
            /*neg_a=*/false, a, /*neg_b=*/false, b,
            /*c_mod=*/(short)0, acc, /*reuse_a=*/false, /*reuse_b=*/false);
    }
    // D layout: VGPR v -> row mbase + v + 8*lh, col nbase + l15
    #pragma unroll
    for (int v = 0; v < 8; ++v) {
        const int row = mbase + v + 8 * lh;
        E[(size_t)row * NROW + nbase + l15] = __expf(1.0f - acc[v]);
    }
}

// ---------------- stage 4: row sums S[i] = 2049*e^2 + sum_t E[i,t] -------
__global__ __launch_bounds__(256) void rowsum_kernel(const float* __restrict__ E,
                                                     float* __restrict__ S) {
    __shared__ float red[8];
    const int i = blockIdx.x, tid = threadIdx.x;
    const float4* row = reinterpret_cast<const float4*>(E + (size_t)i * NROW);
    const float4 x = row[tid * 2], y = row[tid * 2 + 1];
    float s = x.x + x.y + x.z + x.w + y.x + y.y + y.z + y.w;
    #pragma unroll
    for (int d = 16; d; d >>= 1) s += __shfl_xor(s, d, 32);
    if ((tid & 31) == 0) red[tid >> 5] = s;
    __syncthreads();
    if (tid == 0) {
        float t = 0.0f;
        #pragma unroll
        for (int w = 0; w < 8; ++w) t += red[w];
        S[i] = 2049.0f * EXP2F + t;
    }
}

// ---------------- stage 5: serial DP (soft-DTW-like scan) ----------------
// z = -D/gamma. Thread t owns z[1+8t .. 8+8t]; z[0] replicated in all threads.
// Each row: per-thread 8-elem LSE scan, shuffle wave scan, LDS cross-wave.
__device__ __forceinline__ void lse_combine(float& m, float& s, float om, float os) {
    const float M = fmaxf(m, om);
    s = s * __expf(m - M) + os * __expf(om - M);
    m = M;
}

__global__ __launch_bounds__(512) void dp_kernel(const float* __restrict__ E,
                                                 const float* __restrict__ S,
                                                 float* __restrict__ out) {
    __shared__ float sm[16], ss[16];
    const int tid = threadIdx.x;
    const int lane = tid & 31;
    const int wv = tid >> 5;

    float z[8];
    // row 0 init: z = -C[0][j]/gamma ; col j=1+8t+e : odd j (e even) -> cost, even j -> threshold
    {
        const float invs = INV_GAMMA / S[0];
        const float4 ev = *reinterpret_cast<const float4*>(E + tid * 4);
        const float ecol[4] = {ev.x, ev.y, ev.z, ev.w};
        #pragma unroll
        for (int e = 0; e < 8; ++e)
            z[e] = -invs * ((e & 1) ? EXP2F : ecol[e >> 1]);
    }
    float z0 = -INV_GAMMA * EXP2F / S[0];   // column 0 is a pure accumulator

    for (int i = 1; i < NROW; ++i) {
        __builtin_prefetch(E + (size_t)i * NROW + tid * 4, 0, 1);  // global_prefetch next row
        // (1) local inclusive prefix-LSE over the 8 owned elements
        float lm[8], ls[8];
        float cm = z[0], cs = 1.0f;
        lm[0] = cm; ls[0] = cs;
        #pragma unroll
        for (int e = 1; e < 8; ++e) {
            const float M = fmaxf(cm, z[e]);
            cs = cs * __expf(cm - M) + __expf(z[e] - M);
            cm = M;
            lm[e] = cm; ls[e] = cs;
        }
        // (2) wave-level inclusive scan of thread totals (shuffles, no barrier)
        float wm = cm, wsv = cs;
        #pragma unroll
        for (int d = 1; d < 32; d <<= 1) {
            const float om = __shfl_up(wm, d, 32);
            const float os = __shfl_up(wsv, d, 32);
            if (lane >= d) lse_combine(wm, wsv, om, os);
        }
        float em = __shfl_up(wm, 1, 32);     // lane-exclusive
        float es = __shfl_up(wsv, 1, 32);
        if (lane == 0) { em = -1e30f; es = 0.0f; }
        if (lane == 31) { sm[wv] = wm; ss[wv] = wsv; }
        __syncthreads();
        // carry = z0 (+) waveTotals[0..wv-1] (+) lane-exclusive
        float km = z0, ks = 1.0f;
        for (int w = 0; w < wv; ++w) lse_combine(km, ks, sm[w], ss[w]);
        lse_combine(km, ks, em, es);
        __syncthreads();
        // (3) new row: z_new[j] = lcse[idx[j]] - C[i][j]/gamma
        const float invs = INV_GAMMA / S[i];
        const float4 ev = *reinterpret_cast<const float4*>(E + (size_t)i * NROW + tid * 4);
        const float ecol[4] = {ev.x, ev.y, ev.z, ev.w};
        #pragma unroll
        for (int e = 0; e < 8; ++e) {
            float pm = km, psv = ks;           // j = 1+8t+e ; idx = j (even j) or j-1
            if (e & 1) {                       // j even: inclusive through e
                lse_combine(pm, psv, lm[e], ls[e]);
            } else if (e > 0) {                // j odd: inclusive through e-1
                lse_combine(pm, psv, lm[e - 1], ls[e - 1]);
            }                                   // e==0: carry alone (lcse at 8t)
            const float p = pm + __logf(psv);
            z[e] = p - invs * ((e & 1) ? EXP2F : ecol[e >> 1]);
        }
        z0 -= invs * EXP2F;                    // D[i][0] = D[i-1][0] + C[i][0]
    }

    // final: loss = -gamma * LSE(z_last over all 4097)
    float fm = z[0], fs = 1.0f;
    #pragma unroll
    for (int e = 1; e < 8; ++e) {
        const float M = fmaxf(fm, z[e]);
        fs = fs * __expf(fm - M) + __expf(z[e] - M);
        fm = M;
    }
    #pragma unroll
    for (int d = 1; d < 32; d <<= 1) {
        const float om = __shfl_xor(fm, d, 32);
        const float os = __shfl_xor(fs, d, 32);
        lse_combine(fm, fs, om, os);
    }
    if (lane == 0) { sm[wv] = fm; ss[wv] = fs; }
    __syncthreads();
    if (tid == 0) {
        float gm = z0, gs = 1.0f;
        for (int w = 0; w < 16; ++w) lse_combine(gm, gs, sm[w], ss[w]);
        out[0] = -(1.0f / INV_GAMMA) * (gm + __logf(gs));
    }
}

// ---------------- host launch ----------------
extern "C" void kernel_launch(void* const* d_in, const int* in_sizes, int n_in,
                              void* d_out, int out_size, void* d_ws, size_t ws_size,
                              hipStream_t stream) {
    (void)in_sizes; (void)n_in; (void)out_size; (void)ws_size;
    const float* A = (const float*)d_in[0];
    const float* B = (const float*)d_in[1];
    float* out = (float*)d_out;

    // workspace layout (~24 MB)
    float* An = (float*)d_ws;                       // 2048*512
    float* Bn = An + (size_t)NROW * NDIM;           // 2048*512
    float* E  = Bn + (size_t)NROW * NDIM;           // 2048*2048
    float* S  = E + (size_t)NROW * NROW;            // 2048
    int* permA = (int*)(S + NROW);                  // 2048
    int* permB = permA + NROW;                      // 2048

    rank_kernel<<<NROW / 256, 256, 0, stream>>>(A, permA);
    rank_kernel<<<NROW / 256, 256, 0, stream>>>(B, permB);
    norm_permute_kernel<<<NROW, 128, 0, stream>>>(A, permA, An);
    norm_permute_kernel<<<NROW, 128, 0, stream>>>(B, permB, Bn);
    gemm_exp_kernel<<<dim3(NROW / 64, NROW / 32), 256, 0, stream>>>(An, Bn, E);
    rowsum_kernel<<<NROW, 256, 0, stream>>>(E, S);
    dp_kernel<<<1, 512, 0, stream>>>(E, S, out);
}